// KANModel_91001767067678
// MI455X (gfx1250) — compile-verified
//
#include <hip/hip_runtime.h>

typedef __attribute__((ext_vector_type(2))) float v2f;
typedef __attribute__((ext_vector_type(8))) float v8f;

#define T_   52
#define N_   1000
#define D_   2
#define G_   150
#define H_   50
#define T2_  50          // T-2
#define K_   2000        // N*D contraction length
#define GT_  10          // ceil(G/16) g-tiles
#define NSLICE_ 8        // k-slices (waves per block in proj kernel)
#define KSLICE_ 256

// ---------------------------------------------------------------------------
// Kernel A: elementwise scalar MLPs  drift = MLP_d(x), Dh = 0.5*MLP_s(x)
// Also computes dt = (max(t)-min(t))/(T-1) once.
// ---------------------------------------------------------------------------
__global__ void __launch_bounds__(256)
mlp_kernel(const float* __restrict__ TX, const float* __restrict__ t,
           const float* __restrict__ Wd1, const float* __restrict__ bd1,
           const float* __restrict__ Wd2, const float* __restrict__ bd2,
           const float* __restrict__ Ws1, const float* __restrict__ bs1,
           const float* __restrict__ Ws2, const float* __restrict__ bs2,
           float* __restrict__ drift, float* __restrict__ Dh,
           float* __restrict__ dtval) {
    int i = blockIdx.x * blockDim.x + threadIdx.x;
    if (i == 0) {
        float mn = t[0], mx = t[0];
        for (int k = 1; k < T_; ++k) {
            float v = t[k];
            mn = fminf(mn, v);
            mx = fmaxf(mx, v);
        }
        dtval[0] = (mx - mn) / (float)(T_ - 1);
    }
    if (i >= T_ * K_) return;
    float x  = TX[i];
    float ad = 0.0f, as = 0.0f;
#pragma unroll 5
    for (int h = 0; h < H_; ++h) {
        ad += tanhf(fmaf(x, Wd1[h], bd1[h])) * Wd2[h];
        as += tanhf(fmaf(x, Ws1[h], bs1[h])) * Ws2[h];
    }
    drift[i] = ad + bd2[0];
    Dh[i]    = 0.5f * (as + bs2[0]);
}

// ---------------------------------------------------------------------------
// Kernel B: WMMA-based projection.
//   P[gtile][t][slice][m] = partial of
//     sum_k  g1a*drift[t] + g2a*Dh[t] + 4*(g1b*drift[t+1] + g2b*Dh[t+1])
//          + g1c*drift[t+2] + g2c*Dh[t+2]
// One block = (gtile, t); 8 waves, wave w owns k in [w*256, w*256+256).
// A operand = 16 g-rows x 4 k of the big tensor (streamed non-temporally),
// B operand = small vector broadcast to all 16 columns (from LDS) -> every
// accumulator column equals the 16 dot products; column 0 is extracted.
// ---------------------------------------------------------------------------
__global__ void __launch_bounds__(256)
proj_kernel(const float* __restrict__ g1a, const float* __restrict__ g2a,
            const float* __restrict__ g1b, const float* __restrict__ g2b,
            const float* __restrict__ g1c, const float* __restrict__ g2c,
            const float* __restrict__ drift, const float* __restrict__ Dh,
            float* __restrict__ P) {
    __shared__ __align__(16) float sD[3 * K_];   // drift[t..t+2][0..K)
    __shared__ __align__(16) float sH[3 * K_];   // Dh   [t..t+2][0..K)

    const int gt  = blockIdx.x;   // 0..GT_-1
    const int t   = blockIdx.y;   // 0..T2_-1
    const int tid = threadIdx.x;

    for (int idx = tid; idx < 3 * K_; idx += 256) {
        int tt = idx / K_;
        int kk = idx - tt * K_;
        sD[idx] = drift[(t + tt) * K_ + kk];
        sH[idx] = Dh[(t + tt) * K_ + kk];
    }
    __syncthreads();

    const int wave  = tid >> 5;
    const int lane  = tid & 31;
    const int m     = lane & 15;          // A-matrix row (g offset)
    const int halfk = (lane >> 4) << 1;   // lanes 16-31 hold k+2,k+3
    const int g      = gt * 16 + m;
    const bool gval  = (g < G_);
    const long rowbase = ((long)(gval ? g : (G_ - 1)) * T2_ + t) * (long)K_;

    const int k0 = wave * KSLICE_;
    const int k1 = (k0 + KSLICE_ < K_) ? (k0 + KSLICE_) : K_;

    v8f acc0 = {0.f,0.f,0.f,0.f,0.f,0.f,0.f,0.f};   // shift 0 (coef 1)
    v8f acc1 = {0.f,0.f,0.f,0.f,0.f,0.f,0.f,0.f};   // shift 1 (coef 4)
    v8f acc2 = {0.f,0.f,0.f,0.f,0.f,0.f,0.f,0.f};   // shift 2 (coef 1)

    for (int k = k0; k < k1; k += 4) {
        const int kk = k + halfk;
        v2f a, b;

        // ---- term s=0 : gaussn_* with drift[t]/Dh[t] -------------------
        a = __builtin_nontemporal_load((const v2f*)(g1a + rowbase + kk));
        if (!gval) { a.x = 0.f; a.y = 0.f; }
        b = *(const v2f*)(sD + kk);
        acc0 = __builtin_amdgcn_wmma_f32_16x16x4_f32(false, a, false, b,
                                                     (short)0, acc0, false, false);
        a = __builtin_nontemporal_load((const v2f*)(g2a + rowbase + kk));
        if (!gval) { a.x = 0.f; a.y = 0.f; }
        b = *(const v2f*)(sH + kk);
        acc0 = __builtin_amdgcn_wmma_f32_16x16x4_f32(false, a, false, b,
                                                     (short)0, acc0, false, false);

        // ---- term s=1 : gaussnp1_* with drift[t+1]/Dh[t+1] -------------
        a = __builtin_nontemporal_load((const v2f*)(g1b + rowbase + kk));
        if (!gval) { a.x = 0.f; a.y = 0.f; }
        b = *(const v2f*)(sD + K_ + kk);
        acc1 = __builtin_amdgcn_wmma_f32_16x16x4_f32(false, a, false, b,
                                                     (short)0, acc1, false, false);
        a = __builtin_nontemporal_load((const v2f*)(g2b + rowbase + kk));
        if (!gval) { a.x = 0.f; a.y = 0.f; }
        b = *(const v2f*)(sH + K_ + kk);
        acc1 = __builtin_amdgcn_wmma_f32_16x16x4_f32(false, a, false, b,
                                                     (short)0, acc1, false, false);

        // ---- term s=2 : gaussnp2_* with drift[t+2]/Dh[t+2] -------------
        a = __builtin_nontemporal_load((const v2f*)(g1c + rowbase + kk));
        if (!gval) { a.x = 0.f; a.y = 0.f; }
        b = *(const v2f*)(sD + 2 * K_ + kk);
        acc2 = __builtin_amdgcn_wmma_f32_16x16x4_f32(false, a, false, b,
                                                     (short)0, acc2, false, false);
        a = __builtin_nontemporal_load((const v2f*)(g2c + rowbase + kk));
        if (!gval) { a.x = 0.f; a.y = 0.f; }
        b = *(const v2f*)(sH + 2 * K_ + kk);
        acc2 = __builtin_amdgcn_wmma_f32_16x16x4_f32(false, a, false, b,
                                                     (short)0, acc2, false, false);
    }

    // D column 0 lives in lane 0 (rows 0-7, vgprs 0-7) and lane 16 (rows 8-15)
    if (lane == 0 || lane == 16) {
        const int mrow = (lane >> 4) * 8;
        float* dst = P + (((long)gt * T2_ + t) * NSLICE_ + wave) * 16 + mrow;
#pragma unroll
        for (int r = 0; r < 8; ++r)
            dst[r] = acc0[r] + 4.0f * acc1[r] + acc2[r];
    }
}

// ---------------------------------------------------------------------------
// Kernel C: per (g,t): reduce bq = mean_n(gaussnp2_0) - mean_n(gaussn_0),
// sum the 8 k-slice partials, scale by dt/(3N), subtract bq.
// ---------------------------------------------------------------------------
__global__ void __launch_bounds__(128)
finalize_kernel(const float* __restrict__ g0n, const float* __restrict__ g0p2,
                const float* __restrict__ P, const float* __restrict__ dtval,
                float* __restrict__ out) {
    __shared__ float red[128];
    const int b = blockIdx.x;          // 0 .. G_*T2_-1
    const int g = b / T2_;
    const int t = b - g * T2_;
    const long base = ((long)g * T2_ + t) * (long)N_;

    float s = 0.0f;
    for (int n = threadIdx.x; n < N_; n += 128)
        s += g0p2[base + n] - g0n[base + n];
    red[threadIdx.x] = s;
    __syncthreads();
    for (int off = 64; off > 0; off >>= 1) {
        if ((int)threadIdx.x < off) red[threadIdx.x] += red[threadIdx.x + off];
        __syncthreads();
    }
    if (threadIdx.x == 0) {
        float aq = 0.0f;
        const float* p = P + (((long)(g >> 4) * T2_ + t) * NSLICE_) * 16 + (g & 15);
#pragma unroll
        for (int sl = 0; sl < NSLICE_; ++sl) aq += p[sl * 16];
        const float dt = dtval[0];
        out[b] = aq * dt / (3.0f * (float)N_) - red[0] / (float)N_;
    }
}

// ---------------------------------------------------------------------------
extern "C" void kernel_launch(void* const* d_in, const int* in_sizes, int n_in,
                              void* d_out, int out_size, void* d_ws, size_t ws_size,
                              hipStream_t stream) {
    const float* TX         = (const float*)d_in[0];
    const float* t          = (const float*)d_in[1];
    const float* gaussn_0   = (const float*)d_in[2];
    const float* gaussn_1   = (const float*)d_in[3];
    const float* gaussn_2   = (const float*)d_in[4];
    const float* gaussnp1_1 = (const float*)d_in[5];
    const float* gaussnp1_2 = (const float*)d_in[6];
    const float* gaussnp2_0 = (const float*)d_in[7];
    const float* gaussnp2_1 = (const float*)d_in[8];
    const float* gaussnp2_2 = (const float*)d_in[9];
    const float* Wd1 = (const float*)d_in[10];
    const float* bd1 = (const float*)d_in[11];
    const float* Wd2 = (const float*)d_in[12];
    const float* bd2 = (const float*)d_in[13];
    const float* Ws1 = (const float*)d_in[14];
    const float* bs1 = (const float*)d_in[15];
    const float* Ws2 = (const float*)d_in[16];
    const float* bs2 = (const float*)d_in[17];

    float* ws    = (float*)d_ws;
    float* drift = ws;                              // T_*K_ floats
    float* Dh    = drift + (size_t)T_ * K_;         // T_*K_ floats
    float* P     = Dh + (size_t)T_ * K_;            // GT_*T2_*NSLICE_*16 floats
    float* dtv   = P + (size_t)GT_ * T2_ * NSLICE_ * 16;  // 1 float

    mlp_kernel<<<(T_ * K_ + 255) / 256, 256, 0, stream>>>(
        TX, t, Wd1, bd1, Wd2, bd2, Ws1, bs1, Ws2, bs2, drift, Dh, dtv);

    dim3 gridB(GT_, T2_);
    proj_kernel<<<gridB, 256, 0, stream>>>(
        gaussn_1, gaussn_2, gaussnp1_1, gaussnp1_2, gaussnp2_1, gaussnp2_2,
        drift, Dh, P);

    finalize_kernel<<<G_ * T2_, 128, 0, stream>>>(
        gaussn_0, gaussnp2_0, P, dtv, (float*)d_out);
}